// KVCache_29240137351817
// MI455X (gfx1250) — compile-verified
//
#include <hip/hip_runtime.h>
#include <hip/hip_bf16.h>

// Problem constants (from the reference): B,H,L,D,S
#define Bq 4
#define Hq 32
#define Lq 8192   // power of 2 (2^13)
#define Dq 128
#define Sq 2048

static constexpr unsigned long long NROWS = (unsigned long long)Bq * Hq * Lq;    // 1,048,576 rows/tensor
static constexpr unsigned long long NELEM = NROWS * Dq;                          // elements/tensor
static constexpr unsigned CHUNKS_PER_ROW  = (Dq * 2) / 16;                       // 256B row -> 16 x b128
static constexpr unsigned long long CHUNKS_PER_TENSOR = NROWS * CHUNKS_PER_ROW;  // 16,777,216
static constexpr unsigned UNROLL = 8;                   // async chunks per thread
static constexpr unsigned CHUNKS_PER_BLOCK = 256 * UNROLL;  // 2048

// ---------------------------------------------------------------------------
// Kernel 1: inv[l] = -1 for all l in [0,L)
// ---------------------------------------------------------------------------
__global__ void inv_init(int* __restrict__ inv) {
    inv[blockIdx.x * 256u + threadIdx.x] = -1;
}

// ---------------------------------------------------------------------------
// Kernel 2: inv[fill_indices[s]] = s   (fill_indices are unique -> no races)
// ---------------------------------------------------------------------------
__global__ void inv_scatter(const int* __restrict__ fill, int* __restrict__ inv) {
    unsigned s = blockIdx.x * 256u + threadIdx.x;
    inv[fill[s]] = (int)s;
}

// ---------------------------------------------------------------------------
// Kernel 3: pos_new[b,0,l] = inv[l]>=0 ? input_pos[inv[l]] : pos[b,0,l]
// ---------------------------------------------------------------------------
__global__ void pos_fill(const int* __restrict__ pos_in,
                         const int* __restrict__ input_pos,
                         const int* __restrict__ inv,
                         int* __restrict__ out_pos) {
    unsigned i = blockIdx.x * 256u + threadIdx.x;   // i in [0, B*L)
    unsigned l = i & (Lq - 1u);
    int s = inv[l];
    out_pos[i] = (s >= 0) ? input_pos[s] : pos_in[i];
}

// ---------------------------------------------------------------------------
// Kernel 4: fused one-pass scatter-copy for k and v via the CDNA5 async
// global<->LDS DMA path (ASYNCcnt):
//   8x global_load_async_to_lds_b128 (NT) -> s_wait_asynccnt 0 ->
//   8x global_store_async_from_lds_b128 (NT)
// Streaming data is tagged non-temporal so the 1.15 GiB stream does not
// thrash the 192 MB L2; the 32 KB inv table uses default RT policy and
// stays cache-resident. Block = 256 threads = 2048 chunks;
// CHUNKS_PER_TENSOR % 2048 == 0, so each block is entirely in one tensor.
// ---------------------------------------------------------------------------
__global__ __launch_bounds__(256) void kv_scatter_copy(
    const __hip_bfloat16* __restrict__ k_cache,
    const __hip_bfloat16* __restrict__ v_cache,
    const __hip_bfloat16* __restrict__ k_val,
    const __hip_bfloat16* __restrict__ v_val,
    const int* __restrict__ inv,
    __hip_bfloat16* __restrict__ out_k,
    __hip_bfloat16* __restrict__ out_v)
{
    __shared__ char smem[256 * UNROLL * 16];  // 16B staging slot per chunk
    const unsigned tid = threadIdx.x;

    unsigned long long cbase = (unsigned long long)blockIdx.x * CHUNKS_PER_BLOCK;
    const bool second = (cbase >= CHUNKS_PER_TENSOR);
    if (second) cbase -= CHUNKS_PER_TENSOR;
    const char* __restrict__ cache = (const char*)(second ? v_cache : k_cache);
    const char* __restrict__ val   = (const char*)(second ? v_val   : k_val);
    char* __restrict__ out         = (char*)(second ? out_v : out_k);

    // Generic pointer to __shared__ is {SHARED_BASE, lds_offset}; low 32 bits
    // are the LDS byte address the async-LDS instructions consume.
    const unsigned lds_base =
        (unsigned)(unsigned long long)(uintptr_t)(&smem[tid * (UNROLL * 16u)]);

    const char* src[UNROLL];
    unsigned long long dstoff[UNROLL];
#pragma unroll
    for (unsigned j = 0; j < UNROLL; ++j) {
        unsigned long long c = cbase + j * 256u + tid;       // chunk id in tensor
        unsigned chunk         = (unsigned)(c & 15ull);      // chunk within row
        unsigned long long row = c >> 4;                     // [0, B*H*L)
        unsigned l             = (unsigned)(row & (Lq - 1ull));
        unsigned long long bh  = row >> 13;                  // row / L
        int s = inv[l];
        dstoff[j] = row * (unsigned long long)(Dq * 2) + chunk * 16ull;
        if (s >= 0) {
            // filled slot: pull from k_val/v_val at [bh, s, :]
            src[j] = val + (bh * (unsigned long long)Sq + (unsigned)s)
                               * (unsigned long long)(Dq * 2)
                         + chunk * 16ull;
        } else {
            // untouched slot: pass through the old cache row
            src[j] = cache + dstoff[j];
        }
    }

#pragma unroll
    for (unsigned j = 0; j < UNROLL; ++j) {
        asm volatile("global_load_async_to_lds_b128 %0, %1, off th:TH_LOAD_NT"
                     :: "v"(lds_base + j * 16u), "v"(src[j])
                     : "memory");
    }
    // Async LDS writes are unordered w.r.t. the following LDS reads -> wait.
    asm volatile("s_wait_asynccnt 0x0" ::: "memory");
#pragma unroll
    for (unsigned j = 0; j < UNROLL; ++j) {
        asm volatile("global_store_async_from_lds_b128 %0, %1, off th:TH_STORE_NT"
                     :: "v"(out + dstoff[j]), "v"(lds_base + j * 16u)
                     : "memory");
    }
    // Drain stores (S_ENDPGM also implies wait-idle; this makes it explicit).
    asm volatile("s_wait_asynccnt 0x0" ::: "memory");
}

// ---------------------------------------------------------------------------
// Host-side launch (graph-capture safe: kernel launches only).
// Inputs: 0=k_cache 1=v_cache 2=pos 3=fill_indices 4=input_pos 5=k_val 6=v_val
// Output: [k_new bf16 | v_new bf16 | pos_new int32] concatenated.
// ---------------------------------------------------------------------------
extern "C" void kernel_launch(void* const* d_in, const int* in_sizes, int n_in,
                              void* d_out, int out_size, void* d_ws, size_t ws_size,
                              hipStream_t stream) {
    (void)in_sizes; (void)n_in; (void)out_size; (void)ws_size;

    const __hip_bfloat16* k_cache = (const __hip_bfloat16*)d_in[0];
    const __hip_bfloat16* v_cache = (const __hip_bfloat16*)d_in[1];
    const int*            pos_in  = (const int*)d_in[2];
    const int*            fill    = (const int*)d_in[3];
    const int*            ipos    = (const int*)d_in[4];
    const __hip_bfloat16* k_val   = (const __hip_bfloat16*)d_in[5];
    const __hip_bfloat16* v_val   = (const __hip_bfloat16*)d_in[6];

    __hip_bfloat16* out_k   = (__hip_bfloat16*)d_out;
    __hip_bfloat16* out_v   = out_k + NELEM;
    int*            out_pos = (int*)(out_v + NELEM);

    int* inv = (int*)d_ws;  // L x int32 = 32 KB scratch

    inv_init   <<<Lq / 256,        256, 0, stream>>>(inv);
    inv_scatter<<<Sq / 256,        256, 0, stream>>>(fill, inv);
    pos_fill   <<<(Bq * Lq) / 256, 256, 0, stream>>>(pos_in, ipos, inv, out_pos);

    const unsigned nblocks =
        (unsigned)((2ull * CHUNKS_PER_TENSOR) / CHUNKS_PER_BLOCK);  // 16384
    kv_scatter_copy<<<nblocks, 256, 0, stream>>>(k_cache, v_cache, k_val, v_val,
                                                 inv, out_k, out_v);
}